// LensCrackFault_33371895890250
// MI455X (gfx1250) — compile-verified
//
#include <hip/hip_runtime.h>
#include <cstdint>

#define HH 512
#define WW 512
#define CC 3
#define BB 32
#define NLINES 6
#define CRACK_V 0.05f

typedef float v4f __attribute__((ext_vector_type(4)));

// --------------------------------------------------------------------------
// Bulk stream: out = clamp(x, 0, 1).
// 96MB in + 96MB out => HBM-bound, ~8.3us floor @ 23.3 TB/s.
// Stage loads through LDS with CDNA5 async global->LDS 128b copies
// (ASYNCcnt-tracked), 4 independent ops in flight per thread for MLP,
// then clamp in VALU and write with 128-bit non-temporal stores.
// Each lane reads back only its own LDS slots, so a per-wave
// s_wait_asynccnt 0 is sufficient (no workgroup barrier needed).
// --------------------------------------------------------------------------
__global__ __launch_bounds__(256) void lens_copy_clamp(const v4f* __restrict__ x4,
                                                       v4f* __restrict__ o4) {
  __shared__ v4f tile[1024];                 // 16 KB per block
  const unsigned tid  = threadIdx.x;
  const size_t   base = (size_t)blockIdx.x * 1024u;

  // Issue 4 independent async 128-bit global->LDS transfers (ASYNCcnt += 4).
#pragma unroll
  for (int t = 0; t < 4; ++t) {
    // wave-relative LDS byte offset = low 32 bits of the flat shared address
    unsigned lofs = (unsigned)(uintptr_t)(&tile[t * 256 + tid]);
    unsigned long long g =
        (unsigned long long)(uintptr_t)(x4 + base + (size_t)(t * 256 + tid));
    asm volatile("global_load_async_to_lds_b128 %0, %1, off"
                 :: "v"(lofs), "v"(g) : "memory");
  }
  // Wait for all async LDS writes from this wave to land.
  asm volatile("s_wait_asynccnt 0" ::: "memory");

#pragma unroll
  for (int t = 0; t < 4; ++t) {
    v4f v = tile[t * 256 + tid];
    v4f r;
    r.x = __builtin_fminf(__builtin_fmaxf(v.x, 0.0f), 1.0f);
    r.y = __builtin_fminf(__builtin_fmaxf(v.y, 0.0f), 1.0f);
    r.z = __builtin_fminf(__builtin_fmaxf(v.z, 0.0f), 1.0f);
    r.w = __builtin_fminf(__builtin_fmaxf(v.w, 0.0f), 1.0f);
    __builtin_nontemporal_store(r, o4 + base + (size_t)(t * 256 + tid));
  }
}

// --------------------------------------------------------------------------
// Crack raster: since clip(CRACK_V) == CRACK_V, overwriting the clamped copy
// with 0.05 at every line pixel is exactly equivalent to mask-then-clip.
// One thread per (line, channel) = 576 threads. Replicates the reference
// Bresenham scan exactly: emit (cy,cx) for t = 0..nsteps inclusive, then
//   e2 = 2*err; if (e2 > -dy) {err -= dy; cx += sx;}
//               if (e2 <  dx) {err += dx; cy += sy;}
// Endpoints are in [0,512), so all emitted pixels are in-bounds.
// --------------------------------------------------------------------------
__global__ __launch_bounds__(32) void lens_raster(const int* __restrict__ ep,
                                                  float* __restrict__ out) {
  int tid = blockIdx.x * 32 + threadIdx.x;
  if (tid >= BB * NLINES * CC) return;
  int c = tid % CC;
  int l = tid / CC;            // global line index 0..191
  int b = l / NLINES;

  const int* e = ep + l * 4;   // (y0, x0, y1, x1)
  int y0 = e[0], x0 = e[1], y1 = e[2], x1 = e[3];

  int dx = x1 - x0; if (dx < 0) dx = -dx;
  int dy = y1 - y0; if (dy < 0) dy = -dy;
  int sx = (x0 < x1) ? 1 : -1;
  int sy = (y0 < y1) ? 1 : -1;
  int nsteps = dx > dy ? dx : dy;

  int err = dx - dy;
  int cx = x0, cy = y0;
  float* base = out + ((size_t)b * CC + (size_t)c) * (size_t)(HH * WW);

  for (int t = 0; t <= nsteps; ++t) {
    base[cy * WW + cx] = CRACK_V;
    int e2 = 2 * err;
    if (e2 > -dy) { err -= dy; cx += sx; }
    if (e2 <  dx) { err += dx; cy += sy; }
  }
}

extern "C" void kernel_launch(void* const* d_in, const int* in_sizes, int n_in,
                              void* d_out, int out_size, void* d_ws, size_t ws_size,
                              hipStream_t stream) {
  (void)in_sizes; (void)n_in; (void)d_ws; (void)ws_size; (void)out_size;

  const float* x  = (const float*)d_in[0];   // [32,3,512,512] fp32
  const int*   ep = (const int*)d_in[1];     // [32,6,4] int32 (y0,x0,y1,x1)
  float*       out = (float*)d_out;          // [32,3,512,512] fp32

  const int n4     = (BB * CC * HH * WW) / 4;  // 6,291,456 float4
  const int blocks = n4 / 1024;                // 6144 blocks * 16KB tiles

  lens_copy_clamp<<<blocks, 256, 0, stream>>>((const v4f*)x, (v4f*)out);

  const int rthreads = BB * NLINES * CC;       // 576
  lens_raster<<<(rthreads + 31) / 32, 32, 0, stream>>>(ep, out);
}